// RoiPoolingConv_19181323944536
// MI455X (gfx1250) — compile-verified
//
#include <hip/hip_runtime.h>

// TF crop_and_resize (bilinear, extrapolation 0) for:
//   img  [8, 200, 200, 256] f32  (NHWC, channels contiguous)
//   rois [8, 512, 4]        f32  (y1,x1,y2,x2 normalized)
//   out  [8, 512, 7, 7, 256] f32
//
// Memory-bound gather kernel (AI ~0.4 flop/byte): one output cell per wave32,
// wave-uniform scalars via readfirstlane, 128-bit coalesced gathers, NT stores.

typedef float f4 __attribute__((ext_vector_type(4)));

#define IMG_H   200
#define IMG_W   200
#define IMG_C   256
#define CROP    7
#define NCELLS  (8 * 512 * CROP * CROP)   // 200704
#define CELLS_PER_BLOCK 8                 // 8 wave32 per 256-thread block

__global__ __launch_bounds__(256, 4) void roi_crop_resize_kernel(
    const float* __restrict__ img,
    const float* __restrict__ rois,
    float* __restrict__ out)
{
    const int lane = threadIdx.x & 31;
    // cell id is identical across the 32 lanes of a wave -> make it SGPR-resident
    const int cell = __builtin_amdgcn_readfirstlane(
        (int)blockIdx.x * CELLS_PER_BLOCK + (int)(threadIdx.x >> 5));

    // cell = ((b*512 + r)*7 + py)*7 + px
    const int roi = cell / 49;
    const int pp  = cell - roi * 49;
    const int py  = pp / 7;
    const int px  = pp - py * 7;
    const int b   = roi >> 9;             // box_idx = roi / 512

    const f4 rb = ((const f4*)rois)[roi]; // y1, x1, y2, x2
    const float y1v = rb.x, x1v = rb.y, y2v = rb.z, x2v = rb.w;

    // ys = y1*(H-1) + py * ((y2-y1)*(H-1)/(crop-1))   (same assoc. as reference)
    const float ys = y1v * 199.0f + (float)py * ((y2v - y1v) * 199.0f / 6.0f);
    const float xs = x1v * 199.0f + (float)px * ((x2v - x1v) * 199.0f / 6.0f);

    f4* o = (f4*)(out + (size_t)cell * IMG_C);

    const bool valid = (ys >= 0.0f) && (ys <= 199.0f) &&
                       (xs >= 0.0f) && (xs <= 199.0f);
    if (!valid) {                         // wave-uniform branch
        f4 z = (f4){0.0f, 0.0f, 0.0f, 0.0f};
        __builtin_nontemporal_store(z, o + lane);
        __builtin_nontemporal_store(z, o + lane + 32);
        return;
    }

    const float y0f = __builtin_floorf(ys);
    const float x0f = __builtin_floorf(xs);
    const float fy  = ys - y0f;
    const float fx  = xs - x0f;

    int y0 = (int)y0f; y0 = y0 < 0 ? 0 : (y0 > 199 ? 199 : y0);
    int x0 = (int)x0f; x0 = x0 < 0 ? 0 : (x0 > 199 ? 199 : x0);
    const int y1i = (y0 + 1 > 199) ? 199 : y0 + 1;
    const int x1i = (x0 + 1 > 199) ? 199 : x0 + 1;

    // Corner pixel channel vectors (each 256 f32 = 1 KB contiguous)
    const size_t row0 = (size_t)(b * IMG_H + y0 ) * IMG_W;
    const size_t row1 = (size_t)(b * IMG_H + y1i) * IMG_W;
    const f4* p00 = (const f4*)(img + (row0 + x0 ) * IMG_C);
    const f4* p01 = (const f4*)(img + (row0 + x1i) * IMG_C);
    const f4* p10 = (const f4*)(img + (row1 + x0 ) * IMG_C);
    const f4* p11 = (const f4*)(img + (row1 + x1i) * IMG_C);

    // Issue all 8 b128 loads before use: lanes 0..31 cover channels 0..127
    // contiguously (512B/instruction), second load covers 128..255.
    f4 a00 = p00[lane], c00 = p00[lane + 32];
    f4 a01 = p01[lane], c01 = p01[lane + 32];
    f4 a10 = p10[lane], c10 = p10[lane + 32];
    f4 a11 = p11[lane], c11 = p11[lane + 32];

    // Bilinear lerp, same formula/order as reference (fma-contracted by -O3)
    f4 top_a = a00 + (a01 - a00) * fx;
    f4 bot_a = a10 + (a11 - a10) * fx;
    f4 res_a = top_a + (bot_a - top_a) * fy;

    f4 top_b = c00 + (c01 - c00) * fx;
    f4 bot_b = c10 + (c11 - c10) * fx;
    f4 res_b = top_b + (bot_b - top_b) * fy;

    // Output is write-once / never re-read: NT stores keep L2 for gather reuse
    __builtin_nontemporal_store(res_a, o + lane);
    __builtin_nontemporal_store(res_b, o + lane + 32);
}

extern "C" void kernel_launch(void* const* d_in, const int* in_sizes, int n_in,
                              void* d_out, int out_size, void* d_ws, size_t ws_size,
                              hipStream_t stream) {
    const float* img  = (const float*)d_in[0];
    const float* rois = (const float*)d_in[1];
    float* out        = (float*)d_out;

    const int blocks = NCELLS / CELLS_PER_BLOCK;   // 25088, exact
    roi_crop_resize_kernel<<<blocks, 256, 0, stream>>>(img, rois, out);
}